// GATLayer_65532611002920
// MI455X (gfx1250) — compile-verified
//
#include <hip/hip_runtime.h>
#include <math.h>

typedef __attribute__((ext_vector_type(2))) float v2f;
typedef __attribute__((ext_vector_type(8))) float v8f;

#define LDWT 132   // LDS stride (dwords) for transposed W: sWt[col*LDWT + k]
#define LDA  132   // LDS stride for staged A rows
#define NEG_SLOPE 0.2f

__device__ __forceinline__ unsigned ford(float f) {
    unsigned u = __float_as_uint(f);
    return (u & 0x80000000u) ? ~u : (u | 0x80000000u);
}
__device__ __forceinline__ float forddec(unsigned u) {
    return __uint_as_float((u & 0x80000000u) ? (u & 0x7FFFFFFFu) : ~u);
}

// ---------------------------------------------------------------- K0: init
__global__ void init_kernel(float* __restrict__ e_sum, float* __restrict__ cnt,
                            float* __restrict__ denom, unsigned* __restrict__ amax,
                            float* __restrict__ out, const float* __restrict__ bias,
                            int N) {
    const int stride = gridDim.x * blockDim.x;
    const int t0 = blockIdx.x * blockDim.x + threadIdx.x;
    const int NL = N * 128;
    for (int i = t0; i < NL; i += stride) {
        e_sum[i] = 0.0f;
        out[i] = bias[i & 127];
    }
    for (int i = t0; i < N * 4; i += stride) {
        denom[i] = 0.0f;
        amax[i] = 0u;
    }
    for (int i = t0; i < N; i += stride) cnt[i] = 0.0f;
}

// Cooperative load of W (128x128 row-major) into LDS *transposed*:
// sWt[col*LDWT + k] = W[k][col].  128 threads.
__device__ __forceinline__ void load_W_transposed(float* sWt,
                                                  const float* __restrict__ W,
                                                  int tid) {
    for (int i = tid; i < 128 * 32; i += 128) {
        const int row = i >> 5;            // k
        const int c4 = (i & 31) << 2;      // col base
        float4 v = *(const float4*)&W[row * 128 + c4];
        sWt[(c4 + 0) * LDWT + row] = v.x;
        sWt[(c4 + 1) * LDWT + row] = v.y;
        sWt[(c4 + 2) * LDWT + row] = v.z;
        sWt[(c4 + 3) * LDWT + row] = v.w;
    }
}

// ------------------------------------------------- K1: node GEMM (Y = X*W + b)
__global__ void __launch_bounds__(128, 1)
node_gemm_kernel(const float* __restrict__ X, const float* __restrict__ W,
                 const float* __restrict__ b, float* __restrict__ Y, int nTiles) {
    extern __shared__ float smem[];
    float* sWt = smem;                 // 128 * LDWT
    float* sA = smem + 128 * LDWT;     // 4 * 16 * LDA
    const int tid = threadIdx.x;
    const int wave = tid >> 5;
    const int lane = tid & 31;
    const int idx15 = lane & 15;
    const int khalf = (lane >> 4) << 1;  // 0 or 2

    load_W_transposed(sWt, W, tid);

    const int tile = blockIdx.x * 4 + wave;
    const bool valid = tile < nTiles;
    const int rbase = tile * 16;
    float* myA = sA + wave * 16 * LDA;
    if (valid) {
        for (int r = 0; r < 16; ++r) {
            *(float4*)&myA[r * LDA + lane * 4] =
                *(const float4*)&X[(size_t)(rbase + r) * 128 + lane * 4];
        }
    }
    __syncthreads();
    if (!valid) return;

    const int mbase = (lane >> 4) << 3;
    #pragma unroll
    for (int ntp = 0; ntp < 4; ++ntp) {      // pair of 16-col tiles (one head)
        v8f acc0 = {0.f, 0.f, 0.f, 0.f, 0.f, 0.f, 0.f, 0.f};
        v8f acc1 = {0.f, 0.f, 0.f, 0.f, 0.f, 0.f, 0.f, 0.f};
        const int col0 = ntp * 32 + idx15;
        const int col1 = col0 + 16;
        #pragma unroll 4
        for (int k0 = 0; k0 < 32; ++k0) {
            const int k = k0 * 4 + khalf;
            v2f a = *(const v2f*)&myA[idx15 * LDA + k];
            v2f b0 = *(const v2f*)&sWt[col0 * LDWT + k];
            v2f b1 = *(const v2f*)&sWt[col1 * LDWT + k];
            acc0 = __builtin_amdgcn_wmma_f32_16x16x4_f32(false, a, false, b0,
                                                         (short)0, acc0, false, false);
            acc1 = __builtin_amdgcn_wmma_f32_16x16x4_f32(false, a, false, b1,
                                                         (short)0, acc1, false, false);
        }
        const float bv0 = b[col0];
        const float bv1 = b[col1];
        #pragma unroll
        for (int r = 0; r < 8; ++r) {
            const size_t row = (size_t)(rbase + mbase + r) * 128;
            Y[row + col0] = acc0[r] + bv0;
            Y[row + col1] = acc1[r] + bv1;
        }
    }
}

// ---------------- K2: fused edge GEMM + e_sum scatter + attention logits
__global__ void __launch_bounds__(128, 1)
edge_gemm_kernel(const float* __restrict__ EA, const int* __restrict__ EI,
                 const float* __restrict__ W, const float* __restrict__ att,
                 const float* __restrict__ x_l, const float* __restrict__ x_r,
                 float* __restrict__ e_sum, float* __restrict__ cnt,
                 float* __restrict__ alpha, unsigned* __restrict__ amax,
                 int E, int numBlockTiles) {
    extern __shared__ float smem[];
    float* sWt = smem;
    float* sA = smem + 128 * LDWT;
    int* sSrc = (int*)(sA + 4 * 16 * LDA);   // 64 ints
    int* sDst = sSrc + 64;                   // 64 ints
    const int tid = threadIdx.x;
    const int wave = tid >> 5;
    const int lane = tid & 31;
    const int idx15 = lane & 15;
    const int khalf = (lane >> 4) << 1;
    const int mbase = (lane >> 4) << 3;

    load_W_transposed(sWt, W, tid);
    __syncthreads();

    float* myA = sA + wave * 16 * LDA;

    for (int bt = blockIdx.x; bt < numBlockTiles; bt += gridDim.x) {
        const int tile = bt * 4 + wave;       // always < E/16 (uniform grid)
        const int ebase = tile * 16;

        for (int r = 0; r < 16; ++r) {
            *(float4*)&myA[r * LDA + lane * 4] =
                *(const float4*)&EA[(size_t)(ebase + r) * 128 + lane * 4];
        }
        if (lane < 16) sSrc[wave * 16 + lane] = EI[ebase + lane];
        else           sDst[wave * 16 + lane - 16] = EI[E + ebase + lane - 16];
        __syncthreads();

        float part[32];
        #pragma unroll
        for (int i = 0; i < 32; ++i) part[i] = 0.f;

        #pragma unroll
        for (int ntp = 0; ntp < 4; ++ntp) {   // head ntp = cols [32*ntp, 32*ntp+31]
            v8f acc0 = {0.f, 0.f, 0.f, 0.f, 0.f, 0.f, 0.f, 0.f};
            v8f acc1 = {0.f, 0.f, 0.f, 0.f, 0.f, 0.f, 0.f, 0.f};
            const int col0 = ntp * 32 + idx15;
            const int col1 = col0 + 16;
            #pragma unroll 4
            for (int k0 = 0; k0 < 32; ++k0) {
                const int k = k0 * 4 + khalf;
                v2f a = *(const v2f*)&myA[idx15 * LDA + k];
                v2f b0 = *(const v2f*)&sWt[col0 * LDWT + k];
                v2f b1 = *(const v2f*)&sWt[col1 * LDWT + k];
                acc0 = __builtin_amdgcn_wmma_f32_16x16x4_f32(false, a, false, b0,
                                                             (short)0, acc0, false, false);
                acc1 = __builtin_amdgcn_wmma_f32_16x16x4_f32(false, a, false, b1,
                                                             (short)0, acc1, false, false);
            }
            const float av0 = att[col0];
            const float av1 = att[col1];
            #pragma unroll
            for (int r = 0; r < 8; ++r) {
                const int m = mbase + r;
                const int sn = sSrc[wave * 16 + m];
                const int dn = sDst[wave * 16 + m];
                const size_t so = (size_t)sn * 128;
                const size_t dd = (size_t)dn * 128;
                const float ev0 = acc0[r];
                const float ev1 = acc1[r];
                atomicAdd(&e_sum[dd + col0], ev0);
                atomicAdd(&e_sum[dd + col1], ev1);
                float sv0 = x_l[so + col0] + x_r[dd + col0] + ev0;
                float sv1 = x_l[so + col1] + x_r[dd + col1] + ev1;
                sv0 = (sv0 > 0.f) ? sv0 : NEG_SLOPE * sv0;
                sv1 = (sv1 > 0.f) ? sv1 : NEG_SLOPE * sv1;
                part[r * 4 + ntp] += sv0 * av0 + sv1 * av1;
            }
        }
        // reduce over the 16 lanes of each half (rows 0-7 | rows 8-15)
        #pragma unroll
        for (int i = 0; i < 32; ++i) {
            part[i] += __shfl_xor(part[i], 1, 32);
            part[i] += __shfl_xor(part[i], 2, 32);
            part[i] += __shfl_xor(part[i], 4, 32);
            part[i] += __shfl_xor(part[i], 8, 32);
        }
        if (idx15 == 0) {
            for (int r = 0; r < 8; ++r) {
                const int m = mbase + r;
                const int dn = sDst[wave * 16 + m];
                atomicAdd(&cnt[dn], 1.0f);
                const size_t eidx = (size_t)(ebase + m);
                #pragma unroll
                for (int h = 0; h < 4; ++h) {
                    const float lg = part[r * 4 + h];
                    alpha[eidx * 4 + h] = lg;
                    atomicMax(&amax[dn * 4 + h], ford(lg));
                }
            }
        }
        __syncthreads();
    }
}

// -------------------- K3: self-loop scores (entry E+n, src=dst=n, e=mean)
__global__ void self_loop_kernel(const float* __restrict__ e_sum,
                                 const float* __restrict__ cnt,
                                 const float* __restrict__ x_l,
                                 const float* __restrict__ x_r,
                                 const float* __restrict__ att,
                                 float* __restrict__ alpha,
                                 unsigned* __restrict__ amax, int N, int E) {
    const int wave = threadIdx.x >> 5;
    const int lane = threadIdx.x & 31;
    const int node = blockIdx.x * 8 + wave;
    if (node >= N) return;
    const float inv = 1.0f / fmaxf(cnt[node], 1.0f);
    const int k4 = lane * 4;
    float4 el = *(const float4*)&e_sum[(size_t)node * 128 + k4];
    float4 xl = *(const float4*)&x_l[(size_t)node * 128 + k4];
    float4 xr = *(const float4*)&x_r[(size_t)node * 128 + k4];
    float4 at = *(const float4*)&att[k4];
    float p = 0.f;
    {
        float v;
        v = xl.x + xr.x + el.x * inv; v = (v > 0.f) ? v : NEG_SLOPE * v; p += v * at.x;
        v = xl.y + xr.y + el.y * inv; v = (v > 0.f) ? v : NEG_SLOPE * v; p += v * at.y;
        v = xl.z + xr.z + el.z * inv; v = (v > 0.f) ? v : NEG_SLOPE * v; p += v * at.z;
        v = xl.w + xr.w + el.w * inv; v = (v > 0.f) ? v : NEG_SLOPE * v; p += v * at.w;
    }
    p += __shfl_xor(p, 1, 32);
    p += __shfl_xor(p, 2, 32);
    p += __shfl_xor(p, 4, 32);
    if ((lane & 7) == 0) {
        const int h = lane >> 3;
        alpha[(size_t)(E + node) * 4 + h] = p;
        atomicMax(&amax[node * 4 + h], ford(p));
    }
}

// -------------------- K4: ex = exp(logit - amax[dst]); denom[dst] += ex
__global__ void exp_kernel(float* __restrict__ alpha, const int* __restrict__ EI,
                           const unsigned* __restrict__ amax,
                           float* __restrict__ denom, int E, int N) {
    const int i = blockIdx.x * blockDim.x + threadIdx.x;
    if (i >= E + N) return;
    const int dn = (i < E) ? EI[E + i] : (i - E);
    float4 lg = *(float4*)&alpha[(size_t)i * 4];
    uint4 mu = *(const uint4*)&amax[dn * 4];
    float4 ex;
    ex.x = expf(lg.x - forddec(mu.x));
    ex.y = expf(lg.y - forddec(mu.y));
    ex.z = expf(lg.z - forddec(mu.z));
    ex.w = expf(lg.w - forddec(mu.w));
    *(float4*)&alpha[(size_t)i * 4] = ex;
    atomicAdd(&denom[dn * 4 + 0], ex.x);
    atomicAdd(&denom[dn * 4 + 1], ex.y);
    atomicAdd(&denom[dn * 4 + 2], ex.z);
    atomicAdd(&denom[dn * 4 + 3], ex.w);
}

// -------------------- K5: out[dst] += x_l[src] * (ex/denom[dst])  (wave/entry)
__global__ void agg_kernel(const float* __restrict__ alpha,
                           const int* __restrict__ EI,
                           const float* __restrict__ denom,
                           const float* __restrict__ x_l,
                           float* __restrict__ out, int E, int N) {
    const int wid = (blockIdx.x * blockDim.x + threadIdx.x) >> 5;
    if (wid >= E + N) return;
    const int lane = threadIdx.x & 31;
    int sn, dn;
    if (wid < E) { sn = EI[wid]; dn = EI[E + wid]; }
    else         { sn = dn = wid - E; }
    const int h = lane >> 3;
    const float w = alpha[(size_t)wid * 4 + h] / denom[dn * 4 + h];
    const int k4 = lane * 4;
    float4 xv = *(const float4*)&x_l[(size_t)sn * 128 + k4];
    float* ob = &out[(size_t)dn * 128 + k4];
    atomicAdd(ob + 0, xv.x * w);
    atomicAdd(ob + 1, xv.y * w);
    atomicAdd(ob + 2, xv.z * w);
    atomicAdd(ob + 3, xv.w * w);
}

extern "C" void kernel_launch(void* const* d_in, const int* in_sizes, int n_in,
                              void* d_out, int out_size, void* d_ws, size_t ws_size,
                              hipStream_t stream) {
    const float* x    = (const float*)d_in[0];
    const int*   ei   = (const int*)d_in[1];
    const float* ea   = (const float*)d_in[2];
    const float* W_l  = (const float*)d_in[3];
    const float* b_l  = (const float*)d_in[4];
    const float* W_r  = (const float*)d_in[5];
    const float* b_r  = (const float*)d_in[6];
    const float* W_e  = (const float*)d_in[7];
    const float* att  = (const float*)d_in[8];
    const float* bias = (const float*)d_in[9];
    float* out = (float*)d_out;

    const int N = in_sizes[0] / 128;   // 50000
    const int E = in_sizes[1] / 2;     // 800000
    const size_t NL = (size_t)N * 128;

    float* ws    = (float*)d_ws;
    float* x_l   = ws;
    float* x_r   = x_l + NL;
    float* e_sum = x_r + NL;
    float* alpha = e_sum + NL;                       // (E+N)*4
    float* cnt   = alpha + (size_t)(E + N) * 4;      // N
    float* denom = cnt + N;                          // N*4
    unsigned* amax = (unsigned*)(denom + (size_t)N * 4); // N*4

    const size_t shN = (size_t)(128 * LDWT + 4 * 16 * LDA) * sizeof(float);
    const size_t shE = shN + 128 * sizeof(int);

    init_kernel<<<1024, 256, 0, stream>>>(e_sum, cnt, denom, amax, out, bias, N);

    const int nTiles = N / 16;                       // 3125 (exact)
    const int nBlocks = (nTiles + 3) / 4;
    node_gemm_kernel<<<nBlocks, 128, shN, stream>>>(x, W_l, b_l, x_l, nTiles);
    node_gemm_kernel<<<nBlocks, 128, shN, stream>>>(x, W_r, b_r, x_r, nTiles);

    const int eBlockTiles = (E / 16) / 4;            // 12500 (exact)
    edge_gemm_kernel<<<1250, 128, shE, stream>>>(ea, ei, W_e, att, x_l, x_r,
                                                 e_sum, cnt, alpha, amax,
                                                 E, eBlockTiles);

    self_loop_kernel<<<(N + 7) / 8, 256, 0, stream>>>(e_sum, cnt, x_l, x_r, att,
                                                      alpha, amax, N, E);

    const int T = E + N;
    exp_kernel<<<(T + 255) / 256, 256, 0, stream>>>(alpha, ei, amax, denom, E, N);
    agg_kernel<<<((size_t)T * 32 + 255) / 256, 256, 0, stream>>>(alpha, ei, denom,
                                                                 x_l, out, E, N);
}